// InfoNCEWithHardNegatives_73735998537795
// MI455X (gfx1250) — compile-verified
//
#include <hip/hip_runtime.h>
#include <hip/hip_bf16.h>
#include <math.h>

#define TEMP_F 0.07f
#define NEG_INF_F (-1e9f)
#define EPS_NORM 1e-12f

typedef __attribute__((ext_vector_type(2))) float v2f;
typedef __attribute__((ext_vector_type(4))) float v4f;
typedef __attribute__((ext_vector_type(8))) float v8f;

// ---------------- block reduction helpers (works for any power-of-2 blockDim <= 256) ----------------
__device__ __forceinline__ float block_reduce_sum(float v, float* red) {
    const int tid = threadIdx.x;
    red[tid] = v;
    __syncthreads();
    for (int s = blockDim.x >> 1; s > 0; s >>= 1) {
        if (tid < s) red[tid] += red[tid + s];
        __syncthreads();
    }
    float r = red[0];
    __syncthreads();
    return r;
}

__device__ __forceinline__ float block_reduce_max(float v, float* red) {
    const int tid = threadIdx.x;
    red[tid] = v;
    __syncthreads();
    for (int s = blockDim.x >> 1; s > 0; s >>= 1) {
        if (tid < s) red[tid] = fmaxf(red[tid], red[tid + s]);
        __syncthreads();
    }
    float r = red[0];
    __syncthreads();
    return r;
}

// ---------------- kernel 1: normalize text embeddings ----------------
// grid = B blocks, 128 threads; one row of D floats per block, float4 (b128) loads.
__global__ void k_text_norm(const float* __restrict__ text, float* __restrict__ tn, int Dn) {
    __shared__ float red[256];
    const int b = blockIdx.x;
    const float* row = text + (size_t)b * Dn;
    float ss = 0.f;
    for (int d = threadIdx.x * 4; d < Dn; d += blockDim.x * 4) {
        v4f x = *(const v4f*)(row + d);
        ss += x.x * x.x + x.y * x.y + x.z * x.z + x.w * x.w;
    }
    ss = block_reduce_sum(ss, red);
    const float inv = 1.0f / fmaxf(sqrtf(ss), EPS_NORM);
    for (int d = threadIdx.x * 4; d < Dn; d += blockDim.x * 4) {
        v4f x = *(const v4f*)(row + d);
        v4f y = {x.x * inv, x.y * inv, x.z * inv, x.w * inv};
        *(v4f*)(tn + (size_t)b * Dn + d) = y;
    }
}

// ---------------- kernel 2: fused visual pass (the 134MB bandwidth kernel) ----------------
// One block (128 threads, b128 loads) per (b,n) row. Single read of the visual tensor:
//   - sum of squares -> norm
//   - dot with tn[b]  -> sims_in[b,n] = dot/norm/T  (unmasked)
//   - if n == clamp(gt): write normalized GT row + sample_valid flag
__global__ void k_visual_fused(const float* __restrict__ vis, const float* __restrict__ tn,
                               const unsigned char* __restrict__ valid, const int* __restrict__ gti,
                               float* __restrict__ sims_in, float* __restrict__ gtv,
                               float* __restrict__ svalid, int Nn, int Dn) {
    __shared__ float red[256];
    const int n = blockIdx.x;
    const int b = blockIdx.y;
    const float* v = vis + ((size_t)b * Nn + n) * Dn;
    const float* t = tn + (size_t)b * Dn;
    float ss = 0.f, dp = 0.f;
    for (int d = threadIdx.x * 4; d < Dn; d += blockDim.x * 4) {
        v4f x = *(const v4f*)(v + d);
        v4f y = *(const v4f*)(t + d);
        ss += x.x * x.x + x.y * x.y + x.z * x.z + x.w * x.w;
        dp += x.x * y.x + x.y * y.y + x.z * y.z + x.w * y.w;
    }
    ss = block_reduce_sum(ss, red);
    dp = block_reduce_sum(dp, red);
    const float nrm = fmaxf(sqrtf(ss), EPS_NORM);
    if (threadIdx.x == 0)
        sims_in[(size_t)b * Nn + n] = dp / nrm * (1.0f / TEMP_F);

    const int gt = gti[b];
    const int gtc = min(max(gt, 0), Nn - 1);
    if (n == gtc) {
        const float inv = 1.0f / nrm;
        for (int d = threadIdx.x * 4; d < Dn; d += blockDim.x * 4) {
            v4f x = *(const v4f*)(v + d);
            v4f y = {x.x * inv, x.y * inv, x.z * inv, x.w * inv};
            *(v4f*)(gtv + (size_t)b * Dn + d) = y;
        }
        if (threadIdx.x == 0)
            svalid[b] = (gt >= 0 && gt < Nn && valid[(size_t)b * Nn + gtc] != 0) ? 1.0f : 0.0f;
    }
}

// ---------------- kernel 3: S = tn @ gtv^T / T via V_WMMA_F32_16X16X4_F32 ----------------
// wave32: each wave owns one 16x16 output tile; K advances 4 per WMMA.
// A frag (16x4): lanes 0-15 -> M=lane, K={k0,k0+1}; lanes 16-31 -> M=lane-16, K={k0+2,k0+3}.
// B frag (4x16): same addressing with N in place of M (B[k][n] = gtv[n][k]).
// C/D (16x16 f32, 8 VGPRs): element (r + 8*(lane>=16), lane&15).
__global__ void k_batch_gemm_wmma(const float* __restrict__ A, const float* __restrict__ Bm,
                                  float* __restrict__ S, int Bn, int Dn) {
    const int lane = threadIdx.x & 31;
    const int wave = threadIdx.x >> 5;
    const int l    = lane & 15;
    const int half = lane >> 4;
    const int ti = blockIdx.y * 8 + wave;  // tile row
    const int tj = blockIdx.x;             // tile col

    const float* arow = A  + (size_t)(ti * 16 + l) * Dn + half * 2;
    const float* brow = Bm + (size_t)(tj * 16 + l) * Dn + half * 2;

    v8f acc = {};
#pragma unroll 8
    for (int k = 0; k < Dn; k += 4) {
        v2f a = *(const v2f*)(arow + k);
        v2f b = *(const v2f*)(brow + k);
        acc = __builtin_amdgcn_wmma_f32_16x16x4_f32(
            /*neg_a=*/false, a, /*neg_b=*/false, b,
            /*c_mod=*/(short)0, acc, /*reuse_a=*/false, /*reuse_b=*/false);
    }

    const float invT = 1.0f / TEMP_F;
    float* srow = S + (size_t)(ti * 16) * Bn + (size_t)tj * 16 + l;
#pragma unroll
    for (int r = 0; r < 8; ++r) {
        const int m = r + 8 * half;
        srow[(size_t)m * Bn] = acc[r] * invT;
    }
}

// ---------------- kernel 4: per-row masked logsumexp / loss / correctness ----------------
__global__ void k_row_lse(const float* __restrict__ sims_in, const float* __restrict__ sbatch,
                          const unsigned char* __restrict__ valid, const int* __restrict__ gti,
                          const float* __restrict__ svalid, float* __restrict__ lossc,
                          float* __restrict__ corr, int Bn, int Nn) {
    __shared__ float red[256];
    const int b = blockIdx.x;
    const int tid = threadIdx.x;
    const int gt = gti[b];
    const int gtc = min(max(gt, 0), Nn - 1);
    const float pos = sims_in[(size_t)b * Nn + gtc];

    // pass 1: max over masked negatives
    float mx = NEG_INF_F;
    for (int n = tid; n < Nn; n += blockDim.x) {
        const bool m = (valid[(size_t)b * Nn + n] != 0) && (n != gtc);
        mx = fmaxf(mx, m ? sims_in[(size_t)b * Nn + n] : NEG_INF_F);
    }
    for (int j = tid; j < Bn; j += blockDim.x) {
        const bool m = (svalid[j] > 0.5f) && (j != b);
        mx = fmaxf(mx, m ? sbatch[(size_t)b * Bn + j] : NEG_INF_F);
    }
    const float max_neg = block_reduce_max(mx, red);
    const float mval = fmaxf(pos, max_neg);

    // pass 2: sum of exp(x - m); masked entries (NEG_INF) underflow to 0 just like the reference
    float se = 0.f;
    for (int n = tid; n < Nn; n += blockDim.x) {
        const bool m = (valid[(size_t)b * Nn + n] != 0) && (n != gtc);
        se += expf((m ? sims_in[(size_t)b * Nn + n] : NEG_INF_F) - mval);
    }
    for (int j = tid; j < Bn; j += blockDim.x) {
        const bool m = (svalid[j] > 0.5f) && (j != b);
        se += expf((m ? sbatch[(size_t)b * Bn + j] : NEG_INF_F) - mval);
    }
    se = block_reduce_sum(se, red);

    if (tid == 0) {
        const float total = se + expf(pos - mval);
        const float lse = logf(total) + mval;
        const float loss = lse - pos;
        const bool has_neg = max_neg > -5e8f;
        const bool contrib = (svalid[b] > 0.5f) && has_neg;
        lossc[b] = contrib ? loss : 0.0f;
        corr[b]  = (contrib && (pos >= max_neg)) ? 1.0f : 0.0f;
    }
}

// ---------------- kernel 5: final scalar reduction ----------------
__global__ void k_final(const float* __restrict__ svalid, const float* __restrict__ lossc,
                        const float* __restrict__ corr, float* __restrict__ out, int Bn) {
    __shared__ float red[256];
    const int tid = threadIdx.x;
    float sv = 0.f, sl = 0.f, sc = 0.f;
    for (int b = tid; b < Bn; b += blockDim.x) {
        sv += svalid[b];
        sl += lossc[b];
        sc += corr[b];
    }
    sv = block_reduce_sum(sv, red);
    sl = block_reduce_sum(sl, red);
    sc = block_reduce_sum(sc, red);
    if (tid == 0) {
        const float denom = fmaxf(sv, 1.0f);
        out[0] = sl / denom;  // total loss
        out[1] = sc / denom;  // accuracy
    }
}

extern "C" void kernel_launch(void* const* d_in, const int* in_sizes, int n_in,
                              void* d_out, int out_size, void* d_ws, size_t ws_size,
                              hipStream_t stream) {
    const float*         text  = (const float*)d_in[0];          // (B, D)
    const float*         vis   = (const float*)d_in[1];          // (B, N, D)
    const unsigned char* valid = (const unsigned char*)d_in[2];  // (B, N) bool (1B)
    const int*           gti   = (const int*)d_in[3];            // (B,) int32
    float* out = (float*)d_out;                                  // [loss, accuracy]

    const int Bn = in_sizes[3];
    const int Nn = in_sizes[2] / Bn;
    const int Dn = in_sizes[0] / Bn;

    float* ws      = (float*)d_ws;
    float* tn      = ws;                              // B*D   normalized text
    float* gtv     = tn + (size_t)Bn * Dn;            // B*D   normalized GT visual
    float* sims_in = gtv + (size_t)Bn * Dn;           // B*N   raw in-sample sims (/T)
    float* sbatch  = sims_in + (size_t)Bn * Nn;       // B*B   batch sims (/T)
    float* svalid  = sbatch + (size_t)Bn * Bn;        // B     sample_valid flag
    float* lossc   = svalid + Bn;                     // B     per-sample loss contribution
    float* corr    = lossc + Bn;                      // B     per-sample correctness

    // D is a multiple of 512 here (D=512): 128 threads x float4 covers a row exactly.
    k_text_norm<<<Bn, 128, 0, stream>>>(text, tn, Dn);
    k_visual_fused<<<dim3(Nn, Bn), 128, 0, stream>>>(vis, tn, valid, gti, sims_in, gtv, svalid, Nn, Dn);
    // tile grid: (B/16) x (B/16); 8 waves (tiles) per 256-thread block along rows
    k_batch_gemm_wmma<<<dim3(Bn / 16, Bn / 128), 256, 0, stream>>>(tn, gtv, sbatch, Bn, Dn);
    k_row_lse<<<Bn, 256, 0, stream>>>(sims_in, sbatch, valid, gti, svalid, lossc, corr, Bn, Nn);
    k_final<<<1, 256, 0, stream>>>(svalid, lossc, corr, out, Bn);
}